// Actor_70068096468039
// MI455X (gfx1250) — compile-verified
//
#include <hip/hip_runtime.h>
#include <math.h>

// Problem constants (match reference)
#define Nn   50000
#define Ee   400000
#define Ff   20
#define Pp   8
#define Dd   128
#define Hh   8
#define Cc   16
#define FFd  512
#define Ll   2
#define LN_EPS 1e-5f

typedef float v2f __attribute__((ext_vector_type(2)));
typedef float v8f __attribute__((ext_vector_type(8)));

// ---------------------------------------------------------------------------
// fp32 WMMA GEMM: out[M,Nout] = act(A[M,K] @ W[K,Nout] + bias [+ add])
// Each wave owns a 16-row x 64-col tile (4 chained V_WMMA_F32_16X16X4_F32
// accumulators), so one A-fragment load (global_load_b64) feeds 4 WMMAs.
// blockIdx.y selects the 64-col group. M is a multiple of 16 (50000=3125*16)
// and Nout a multiple of 64, so EXEC stays all-ones inside every WMMA.
// ---------------------------------------------------------------------------
__global__ void gt_gemm_wmma_f32(const float* __restrict__ A,
                                 const float* __restrict__ W,
                                 const float* __restrict__ bias,
                                 const float* __restrict__ add,   // may be null
                                 float* __restrict__ out,
                                 int M, int K, int Nout, int relu)
{
    const int wave = threadIdx.x >> 5;
    const int lane = threadIdx.x & 31;
    const int row0 = (blockIdx.x * 4 + wave) * 16;
    if (row0 >= M) return;                 // wave-uniform: EXEC stays full
    const int col0 = blockIdx.y * 64;

    const int lhalf = lane >> 4;           // 0: K pair {0,1}, 1: K pair {2,3}
    const int lmod  = lane & 15;           // M (for A) / N (for B,C,D) index

    const float* Arow = A + (size_t)(row0 + lmod) * K;
    const float* Wcol = W + col0 + lmod;

    v8f c0 = {}, c1 = {}, c2 = {}, c3 = {};
    #pragma unroll 4
    for (int k0 = 0; k0 < K; k0 += 4) {
        const int ka = k0 + lhalf * 2;
        const v2f a = *(const v2f*)(Arow + ka);          // single b64 load
        const float* Wp = Wcol + (size_t)ka * Nout;
        v2f b0, b1, b2, b3;
        b0.x = Wp[0];   b0.y = Wp[Nout];
        b1.x = Wp[16];  b1.y = Wp[Nout + 16];
        b2.x = Wp[32];  b2.y = Wp[Nout + 32];
        b3.x = Wp[48];  b3.y = Wp[Nout + 48];
        if (k0 + 64 < K)                                  // uniform guard
            __builtin_prefetch(Arow + ka + 64, 0, 0);     // global_prefetch_b8
        c0 = __builtin_amdgcn_wmma_f32_16x16x4_f32(false, a, false, b0, (short)0, c0, false, false);
        c1 = __builtin_amdgcn_wmma_f32_16x16x4_f32(false, a, false, b1, (short)0, c1, false, false);
        c2 = __builtin_amdgcn_wmma_f32_16x16x4_f32(false, a, false, b2, (short)0, c2, false, false);
        c3 = __builtin_amdgcn_wmma_f32_16x16x4_f32(false, a, false, b3, (short)0, c3, false, false);
    }

    const v8f cs[4] = { c0, c1, c2, c3 };
    #pragma unroll
    for (int j = 0; j < 4; ++j) {
        const int col = col0 + j * 16 + lmod;
        const float bcol = bias[col];
        #pragma unroll
        for (int r = 0; r < 8; ++r) {
            const int row = row0 + r + lhalf * 8;
            float val = cs[j][r] + bcol;
            if (add)  val += add[(size_t)row * Nout + col];
            if (relu) val = fmaxf(val, 0.0f);
            out[(size_t)row * Nout + col] = val;
        }
    }
}

// ---------------------------------------------------------------------------
// Input projections: inp = fa@W_in + b_in ; x = npa@W_emb + b_emb
// ---------------------------------------------------------------------------
__global__ void gt_embed(const float* __restrict__ fa, const float* __restrict__ npa,
                         const float* __restrict__ Win, const float* __restrict__ bin,
                         const float* __restrict__ Wemb, const float* __restrict__ bemb,
                         float* __restrict__ inp, float* __restrict__ x)
{
    int t = blockIdx.x * blockDim.x + threadIdx.x;
    if (t >= Nn * Dd) return;
    const int n = t >> 7, d = t & 127;
    float a = bin[d];
    #pragma unroll
    for (int p = 0; p < Ff; ++p) a += fa[n * Ff + p] * Win[p * Dd + d];
    inp[t] = a;
    float b = bemb[d];
    #pragma unroll
    for (int p = 0; p < Pp; ++p) b += npa[n * Pp + p] * Wemb[p * Dd + d];
    x[t] = b;
}

__global__ void gt_add_inplace(float* __restrict__ x, const float* __restrict__ y, int n)
{
    int t = blockIdx.x * blockDim.x + threadIdx.x;
    if (t < n) x[t] += y[t];
}

__global__ void gt_fill(float* __restrict__ p, float v, int n)
{
    int t = blockIdx.x * blockDim.x + threadIdx.x;
    if (t < n) p[t] = v;
}

// Recompute edge embedding element d (d = h*16+c) from the P=8 edge features.
__device__ __forceinline__ float gt_edge_emb(const float ep[Pp],
                                             const float* __restrict__ WeL, int d)
{
    float s = 0.0f;
    #pragma unroll
    for (int p = 0; p < Pp; ++p) s += ep[p] * WeL[p * Dd + d];
    return s;
}

// ---------------------------------------------------------------------------
// alpha[e,h] = 0.25 * <q[dst,h,:], k[src,h,:] + e[h,:]>; atomic-max into m.
// ---------------------------------------------------------------------------
__global__ void gt_alpha(const float* __restrict__ q, const float* __restrict__ k,
                         const float* __restrict__ epa, const float* __restrict__ WeL,
                         const int* __restrict__ ei,
                         float* __restrict__ alpha, float* __restrict__ m)
{
    int t = blockIdx.x * blockDim.x + threadIdx.x;
    if (t >= Ee * Hh) return;
    const int e = t >> 3, h = t & 7;
    const int src = ei[e], dst = ei[Ee + e];
    float ep[Pp];
    #pragma unroll
    for (int p = 0; p < Pp; ++p) ep[p] = epa[e * Pp + p];
    float acc = 0.0f;
    #pragma unroll
    for (int c = 0; c < Cc; ++c) {
        const int d = h * Cc + c;
        const float kj = k[(size_t)src * Dd + d] + gt_edge_emb(ep, WeL, d);
        acc += q[(size_t)dst * Dd + d] * kj;
    }
    acc *= 0.25f;                          // 1/sqrt(C), C = 16
    alpha[t] = acc;
    atomicMax(&m[dst * Hh + h], acc);      // float max -> global_atomic_max_num_f32
}

// ex = exp(alpha - m[dst]); atomic-add into s; overwrite alpha with ex.
__global__ void gt_exps(float* __restrict__ alpha, const float* __restrict__ m,
                        float* __restrict__ s, const int* __restrict__ ei)
{
    int t = blockIdx.x * blockDim.x + threadIdx.x;
    if (t >= Ee * Hh) return;
    const int e = t >> 3, h = t & 7;
    const int dst = ei[Ee + e];
    const float ex = __expf(alpha[t] - m[dst * Hh + h]);
    alpha[t] = ex;
    atomicAdd(&s[dst * Hh + h], ex);
}

// msg scatter: agg[dst,h,:] += (v[src,h,:] + e[h,:]) * (ex / s[dst,h])
__global__ void gt_scatter(const float* __restrict__ v, const float* __restrict__ epa,
                           const float* __restrict__ WeL, const int* __restrict__ ei,
                           const float* __restrict__ ex, const float* __restrict__ s,
                           float* __restrict__ agg)
{
    int t = blockIdx.x * blockDim.x + threadIdx.x;
    if (t >= Ee * Hh) return;
    const int e = t >> 3, h = t & 7;
    const int src = ei[e], dst = ei[Ee + e];
    const float a = ex[t] / s[dst * Hh + h];
    float ep[Pp];
    #pragma unroll
    for (int p = 0; p < Pp; ++p) ep[p] = epa[e * Pp + p];
    #pragma unroll
    for (int c = 0; c < Cc; ++c) {
        const int d = h * Cc + c;
        const float msg = (v[(size_t)src * Dd + d] + gt_edge_emb(ep, WeL, d)) * a;
        atomicAdd(&agg[(size_t)dst * Dd + d], msg);
    }
}

// ---------------------------------------------------------------------------
// out[n,:] = LayerNorm(a[n,:] + b[n,:]) * g + beta ; one 128-thread block/node
// ---------------------------------------------------------------------------
__global__ void gt_ln(const float* __restrict__ a, const float* __restrict__ b,
                      const float* __restrict__ g, const float* __restrict__ beta,
                      float* __restrict__ out)
{
    __shared__ float red[4];
    const int n = blockIdx.x, d = threadIdx.x;
    const size_t idx = (size_t)n * Dd + d;
    const float val = a[idx] + b[idx];

    float sum = val;
    #pragma unroll
    for (int off = 16; off > 0; off >>= 1) sum += __shfl_down(sum, off, 32);
    if ((d & 31) == 0) red[d >> 5] = sum;
    __syncthreads();
    const float mean = (red[0] + red[1] + red[2] + red[3]) * (1.0f / Dd);
    __syncthreads();

    const float dv = val - mean;
    float vs = dv * dv;
    #pragma unroll
    for (int off = 16; off > 0; off >>= 1) vs += __shfl_down(vs, off, 32);
    if ((d & 31) == 0) red[d >> 5] = vs;
    __syncthreads();
    const float var = (red[0] + red[1] + red[2] + red[3]) * (1.0f / Dd);

    out[idx] = dv * rsqrtf(var + LN_EPS) * g[d] + beta[d];
}

// Output head + allocation mask
__global__ void gt_out(const float* __restrict__ x, const float* __restrict__ Wout,
                       const float* __restrict__ bout, const float* __restrict__ fa,
                       float* __restrict__ out)
{
    int t = blockIdx.x * blockDim.x + threadIdx.x;
    if (t >= Nn * Ff) return;
    const int n = t / Ff, f = t % Ff;
    float acc = bout[f];
    #pragma unroll 8
    for (int d = 0; d < Dd; ++d) acc += x[(size_t)n * Dd + d] * Wout[d * Ff + f];
    float rs = 0.0f;
    #pragma unroll
    for (int p = 0; p < Ff; ++p) rs += fa[n * Ff + p];
    out[t] = (rs < 1.0f) ? acc : -INFINITY;
}

// ---------------------------------------------------------------------------
extern "C" void kernel_launch(void* const* d_in, const int* in_sizes, int n_in,
                              void* d_out, int out_size, void* d_ws, size_t ws_size,
                              hipStream_t stream)
{
    const float* fa    = (const float*)d_in[0];
    const float* npa   = (const float*)d_in[1];
    const float* epa   = (const float*)d_in[2];
    const int*   ei    = (const int*)  d_in[3];
    /* d_in[4] = ptr : unused by the reference computation */
    const float* Win   = (const float*)d_in[5];
    const float* bin   = (const float*)d_in[6];
    const float* Wemb  = (const float*)d_in[7];
    const float* bemb  = (const float*)d_in[8];
    const float* Wq    = (const float*)d_in[9];
    const float* bq    = (const float*)d_in[10];
    const float* Wk    = (const float*)d_in[11];
    const float* bk    = (const float*)d_in[12];
    const float* Wv    = (const float*)d_in[13];
    const float* bv    = (const float*)d_in[14];
    const float* We    = (const float*)d_in[15];
    const float* Wskip = (const float*)d_in[16];
    const float* bskip = (const float*)d_in[17];
    const float* W1    = (const float*)d_in[18];
    const float* b1    = (const float*)d_in[19];
    const float* W2    = (const float*)d_in[20];
    const float* b2    = (const float*)d_in[21];
    const float* ln1g  = (const float*)d_in[22];
    const float* ln1b  = (const float*)d_in[23];
    const float* ln2g  = (const float*)d_in[24];
    const float* ln2b  = (const float*)d_in[25];
    const float* Wout  = (const float*)d_in[26];
    const float* bout  = (const float*)d_in[27];

    float* ws = (float*)d_ws;
    size_t o = 0;
    float* inp   = ws + o; o += (size_t)Nn * Dd;
    float* x     = ws + o; o += (size_t)Nn * Dd;
    float* q     = ws + o; o += (size_t)Nn * Dd;
    float* k     = ws + o; o += (size_t)Nn * Dd;
    float* v     = ws + o; o += (size_t)Nn * Dd;
    float* xtmp  = ws + o; o += (size_t)Nn * Dd;
    float* agg   = ws + o; o += (size_t)Nn * Dd;
    float* hff   = ws + o; o += (size_t)Nn * FFd;
    float* mbuf  = ws + o; o += (size_t)Nn * Hh;
    float* sbuf  = ws + o; o += (size_t)Nn * Hh;
    float* alpha = ws + o; o += (size_t)Ee * Hh;

    const int TB = 256;
    const int gND  = (Nn * Dd + TB - 1) / TB;   // elementwise over [N,D]
    const int gNH  = (Nn * Hh + TB - 1) / TB;
    const int gEH  = (Ee * Hh + TB - 1) / TB;
    const int gNF  = (Nn * Ff + TB - 1) / TB;
    const int gRows = ((Nn / 16) + 3) / 4;      // 3125 row-tiles, 4 waves/block
    const dim3 gemmD(gRows, Dd  / 64);          // Nout = 128 -> 2 col groups
    const dim3 gemmF(gRows, FFd / 64);          // Nout = 512 -> 8 col groups

    gt_embed<<<gND, TB, 0, stream>>>(fa, npa, Win, bin, Wemb, bemb, inp, x);

    for (int l = 0; l < Ll; ++l) {
        const float* WqL = Wq + (size_t)l * Dd * Dd;   const float* bqL = bq + l * Dd;
        const float* WkL = Wk + (size_t)l * Dd * Dd;   const float* bkL = bk + l * Dd;
        const float* WvL = Wv + (size_t)l * Dd * Dd;   const float* bvL = bv + l * Dd;
        const float* WeL = We + (size_t)l * Pp * Dd;
        const float* WsL = Wskip + (size_t)l * Dd * Dd; const float* bsL = bskip + l * Dd;
        const float* W1L = W1 + (size_t)l * Dd * FFd;  const float* b1L = b1 + l * FFd;
        const float* W2L = W2 + (size_t)l * FFd * Dd;  const float* b2L = b2 + l * Dd;

        // x = x + inp
        gt_add_inplace<<<gND, TB, 0, stream>>>(x, inp, Nn * Dd);

        // Q, K, V projections (WMMA)
        gt_gemm_wmma_f32<<<gemmD, 128, 0, stream>>>(x, WqL, bqL, nullptr, q, Nn, Dd, Dd, 0);
        gt_gemm_wmma_f32<<<gemmD, 128, 0, stream>>>(x, WkL, bkL, nullptr, k, Nn, Dd, Dd, 0);
        gt_gemm_wmma_f32<<<gemmD, 128, 0, stream>>>(x, WvL, bvL, nullptr, v, Nn, Dd, Dd, 0);

        // segment softmax state
        gt_fill<<<gNH, TB, 0, stream>>>(mbuf, -INFINITY, Nn * Hh);
        gt_fill<<<gNH, TB, 0, stream>>>(sbuf, 0.0f, Nn * Hh);
        gt_fill<<<gND, TB, 0, stream>>>(agg, 0.0f, Nn * Dd);

        gt_alpha  <<<gEH, TB, 0, stream>>>(q, k, epa, WeL, ei, alpha, mbuf);
        gt_exps   <<<gEH, TB, 0, stream>>>(alpha, mbuf, sbuf, ei);
        gt_scatter<<<gEH, TB, 0, stream>>>(v, epa, WeL, ei, alpha, sbuf, agg);

        // x2 = agg + x @ Wskip + bskip ; x = LN1(x + x2)
        gt_gemm_wmma_f32<<<gemmD, 128, 0, stream>>>(x, WsL, bsL, agg, xtmp, Nn, Dd, Dd, 0);
        gt_ln<<<Nn, Dd, 0, stream>>>(x, xtmp, ln1g + l * Dd, ln1b + l * Dd, x);

        // FFN: x = LN2(x + relu(x@W1+b1)@W2 + b2)
        gt_gemm_wmma_f32<<<gemmF, 128, 0, stream>>>(x, W1L, b1L, nullptr, hff, Nn, Dd, FFd, 1);
        gt_gemm_wmma_f32<<<gemmD, 128, 0, stream>>>(hff, W2L, b2L, nullptr, xtmp, Nn, FFd, Dd, 0);
        gt_ln<<<Nn, Dd, 0, stream>>>(x, xtmp, ln2g + l * Dd, ln2b + l * Dd, x);
    }

    gt_out<<<gNF, TB, 0, stream>>>(x, Wout, bout, fa, (float*)d_out);
}